// EdgePreservationLoss_47021301956632
// MI455X (gfx1250) — compile-verified
//
#include <hip/hip_runtime.h>
#include <stdint.h>

#define HH 512
#define WW 512
#define CHN 3
#define BATCH 32
#define RPB 16                       // interior rows per block
#define NBLOCKS (BATCH * (HH / RPB)) // 1024
#define NTHREADS 256
#define GSTRIDE (WW + 2)             // 514, zero-padded gray rows
#define IMGSTRIDE (CHN * WW)         // 1536 floats per (2-image) staging sub-block

typedef int v4i __attribute__((__vector_size__(16)));
typedef __attribute__((address_space(1))) v4i glob_v4i;
typedef __attribute__((address_space(3))) v4i lds_v4i;

// ---- CDNA5 async global->LDS DMA (ASYNCcnt path) ----------------------------
#if __has_builtin(__builtin_amdgcn_global_load_async_to_lds_b128)
__device__ __forceinline__ void async_copy16(const float* g, float* l) {
    __builtin_amdgcn_global_load_async_to_lds_b128(
        (glob_v4i*)(uintptr_t)g,
        (lds_v4i*)(uintptr_t)(uint32_t)(uintptr_t)l,   // flat->LDS: low 32 bits
        0, 0);
}
#else
#warning "builtin global_load_async_to_lds_b128 missing: using inline-asm path"
__device__ __forceinline__ void async_copy16(const float* g, float* l) {
    uint32_t loff = (uint32_t)(uintptr_t)l;      // LDS byte offset
    uint64_t gaddr = (uint64_t)(uintptr_t)g;
    asm volatile("global_load_async_to_lds_b128 %0, %1, off"
                 :: "v"(loff), "v"(gaddr) : "memory");
}
#endif

#if __has_builtin(__builtin_amdgcn_s_wait_asynccnt)
__device__ __forceinline__ void wait_async_le3() { __builtin_amdgcn_s_wait_asynccnt(3); }
__device__ __forceinline__ void wait_async_le0() { __builtin_amdgcn_s_wait_asynccnt(0); }
#else
#warning "builtin s_wait_asynccnt missing: using inline-asm path"
__device__ __forceinline__ void wait_async_le3() { asm volatile("s_wait_asynccnt 3" ::: "memory"); }
__device__ __forceinline__ void wait_async_le0() { asm volatile("s_wait_asynccnt 0" ::: "memory"); }
#endif
// -----------------------------------------------------------------------------

__global__ __launch_bounds__(NTHREADS)
void edge_loss_partial(const float* __restrict__ pred,
                       const float* __restrict__ targ,
                       float* __restrict__ partial)
{
    // double-buffered raw staging: [buf][im*1536 + ch*512 + col]
    __shared__ float stage[2][2 * IMGSTRIDE];      // 24576 B
    // circular gray rows: [(im*3 + slot)*514 + col+1], cols 0 & 513 are zero pad
    __shared__ float gray[2 * 3 * GSTRIDE];        // 12336 B
    __shared__ float red[NTHREADS];                //  1024 B

    const int t  = threadIdx.x;
    const int bk = blockIdx.x;
    const int b  = bk >> 5;        // batch
    const int rb = bk & 31;        // row-block
    const int r0 = rb * RPB;

    // zero the padded border columns of all 6 gray rows (written once)
    if (t < 12) {
        int row = t >> 1;
        gray[row * GSTRIDE + ((t & 1) ? (GSTRIDE - 1) : 0)] = 0.0f;
    }

    // per-thread DMA chunk descriptors: 3 x float4 per row-group
    const float* cbase[3];
    int coff[3];
#pragma unroll
    for (int i = 0; i < 3; ++i) {
        int f   = (t + i * NTHREADS) * 4;          // flat float index 0..3068
        int im  = f / IMGSTRIDE;                   // 0=pred 1=target
        int rem = f - im * IMGSTRIDE;
        int ch  = rem / WW;
        int col = rem - ch * WW;
        const float* img = im ? targ : pred;
        cbase[i] = img + (size_t)(b * CHN + ch) * HH * WW + col;
        coff[i]  = f;
    }

    auto issue_row = [&](int k) {
        int g  = r0 - 1 + k;
        int gc = g < 0 ? 0 : (g >= HH ? HH - 1 : g);   // clamp; invalid rows zeroed later
#pragma unroll
        for (int i = 0; i < 3; ++i)
            async_copy16(cbase[i] + (size_t)gc * WW, &stage[k & 1][coff[i]]);
    };

    issue_row(0);
    issue_row(1);

    float acc = 0.0f;

    for (int k = 0; k < RPB + 2; ++k) {
        if (k == RPB + 1) wait_async_le0(); else wait_async_le3();
        __syncthreads();                               // row k staged for all waves

        // luma for loop-row k (global row r0-1+k) -> gray slot k%3
        const int g     = r0 - 1 + k;
        const int slot  = k % 3;
        const bool ok   = (g >= 0) && (g < HH);
        const float* sb = stage[k & 1];
#pragma unroll
        for (int im = 0; im < 2; ++im) {
#pragma unroll
            for (int h = 0; h < 2; ++h) {
                int c = t + h * NTHREADS;
                float gy = 0.0f;
                if (ok) {
                    float r  = sb[im * IMGSTRIDE + 0 * WW + c];
                    float gg = sb[im * IMGSTRIDE + 1 * WW + c];
                    float bl = sb[im * IMGSTRIDE + 2 * WW + c];
                    gy = 0.299f * r + 0.587f * gg + 0.114f * bl;
                }
                gray[(im * 3 + slot) * GSTRIDE + c + 1] = gy;
            }
        }
        __syncthreads();                               // gray row visible; staging reads done

        if (k + 2 < RPB + 2) issue_row(k + 2);         // DMA overlaps Sobel below

        if (k >= 2) {
            const int st = (k - 2) % 3, sm = (k - 1) % 3, sbt = k % 3;
#pragma unroll
            for (int h = 0; h < 2; ++h) {
                int cc = t + h * NTHREADS + 1;
                float e[2];
#pragma unroll
                for (int im = 0; im < 2; ++im) {
                    const float* rt  = &gray[(im * 3 + st)  * GSTRIDE];
                    const float* rm  = &gray[(im * 3 + sm)  * GSTRIDE];
                    const float* rbm = &gray[(im * 3 + sbt) * GSTRIDE];
                    float tl = rt[cc - 1],  tc = rt[cc],  tr = rt[cc + 1];
                    float ml = rm[cc - 1],                mr = rm[cc + 1];
                    float bl = rbm[cc - 1], bc = rbm[cc], br = rbm[cc + 1];
                    float ex = (tr - tl) + 2.0f * (mr - ml) + (br - bl);
                    float ey = (bl - tl) + 2.0f * (bc - tc) + (br - tr);
                    e[im] = sqrtf(ex * ex + ey * ey);
                }
                float d = e[0] - e[1];
                acc += d * d;
            }
        }
    }

    // block reduction -> one partial per block (deterministic)
    red[t] = acc;
    __syncthreads();
    for (int s = NTHREADS / 2; s > 0; s >>= 1) {
        if (t < s) red[t] += red[t + s];
        __syncthreads();
    }
    if (t == 0) partial[bk] = red[0];
}

__global__ __launch_bounds__(NTHREADS)
void edge_loss_final(const float* __restrict__ partial, float* __restrict__ out)
{
    __shared__ float red[NTHREADS];
    int t = threadIdx.x;
    float s = 0.0f;
    for (int i = t; i < NBLOCKS; i += NTHREADS) s += partial[i];
    red[t] = s;
    __syncthreads();
    for (int k = NTHREADS / 2; k > 0; k >>= 1) {
        if (t < k) red[t] += red[t + k];
        __syncthreads();
    }
    if (t == 0) out[0] = red[0] * (1.0f / (float)(BATCH * HH * WW));
}

extern "C" void kernel_launch(void* const* d_in, const int* in_sizes, int n_in,
                              void* d_out, int out_size, void* d_ws, size_t ws_size,
                              hipStream_t stream)
{
    const float* pred = (const float*)d_in[0];
    const float* targ = (const float*)d_in[1];
    float* partial = (float*)d_ws;   // NBLOCKS floats = 4 KB scratch

    edge_loss_partial<<<NBLOCKS, NTHREADS, 0, stream>>>(pred, targ, partial);
    edge_loss_final<<<1, NTHREADS, 0, stream>>>(partial, (float*)d_out);
}